// MSDeformAttn_81028853007061
// MI455X (gfx1250) — compile-verified
//
#include <hip/hip_runtime.h>
#include <cstddef>

// ---------------------------------------------------------------------------
// MSDeformAttn for MI455X (gfx1250, wave32).
// N=8, Lq=Lin=5440, d=256, H=8, Ch=32, L=4 levels (64^2,32^2,16^2,8^2 ->
// starts 0,4096,5120,5376), P=4.
// GEMMs: V_WMMA_F32_16X16X4_F32 (exact fp32), 16x64 tile per wave.
// Sampling: one wave per (n,q,h), branchless bilinear, coalesced 128B gathers
// from the L2-resident permuted (n,h,lin,32) value tensor.
// ---------------------------------------------------------------------------

typedef float v2f __attribute__((ext_vector_type(2)));
typedef float v8f __attribute__((ext_vector_type(8)));

#define KDIM   256
#define MROWS  43520      // N * Lq
#define LQ     5440
#define NHEADS 8
#define CHN    32

enum { EPI_NONE = 0, EPI_VALUE = 1, EPI_SOFTMAX = 2 };

__device__ __forceinline__ v8f wmma_f32(v2f a, v2f b, v8f c) {
    // 8 args: (neg_a, A, neg_b, B, c_mod, C, reuse_a, reuse_b)
    return __builtin_amdgcn_wmma_f32_16x16x4_f32(false, a, false, b,
                                                 (short)0, c, false, false);
}

// Softmax over the 16 columns of a 16x16 C tile. C-layout: VGPR i holds row
// (m0+i) on lanes 0-15 and row (m0+i+8) on lanes 16-31, column = lane&15,
// so a width-16 shuffle reduction is exactly a per-row reduction.
__device__ __forceinline__ v8f softmax16(v8f c) {
    v8f r;
#pragma unroll
    for (int i = 0; i < 8; ++i) {
        float v = c[i];
        float mx = v;
#pragma unroll
        for (int d = 1; d < 16; d <<= 1) mx = fmaxf(mx, __shfl_xor(mx, d, 16));
        float e = __expf(v - mx);
        float s = e;
#pragma unroll
        for (int d = 1; d < 16; d <<= 1) s += __shfl_xor(s, d, 16);
        r[i] = e / s;
    }
    return r;
}

// C[M x NC] = A[M x 256] * Wt[NC x 256]^T + bias.
// Block = 256 threads = 8 waves; each wave computes one 16x64 tile
// (4 independent f32 WMMA accumulator chains).
template <int NC, int EPI>
__global__ __launch_bounds__(256, 2)
void gemm_wmma_f32(const float* __restrict__ A,
                   const float* __restrict__ Wt,
                   const float* __restrict__ bias,
                   float* __restrict__ out)
{
    const int lane = threadIdx.x & 31;
    const int wave = threadIdx.x >> 5;
    const int m0   = (blockIdx.x * 8 + wave) * 16;
    const int nb   = blockIdx.y * 64;
    const int r16  = lane & 15;
    const int hi   = lane >> 4;   // 0: K={0,1}, 1: K={2,3} (ISA 16x4 f32 layout)
    const int koff = hi << 1;

    v8f c0, c1, c2, c3;
    const float bb0 = bias[nb +  0 + r16];
    const float bb1 = bias[nb + 16 + r16];
    const float bb2 = bias[nb + 32 + r16];
    const float bb3 = bias[nb + 48 + r16];
#pragma unroll
    for (int i = 0; i < 8; ++i) { c0[i] = bb0; c1[i] = bb1; c2[i] = bb2; c3[i] = bb3; }

    const float* Ap  = A  + (size_t)(m0 + r16) * KDIM + koff;
    const float* Wp0 = Wt + (size_t)(nb +  0 + r16) * KDIM + koff;
    const float* Wp1 = Wt + (size_t)(nb + 16 + r16) * KDIM + koff;
    const float* Wp2 = Wt + (size_t)(nb + 32 + r16) * KDIM + koff;
    const float* Wp3 = Wt + (size_t)(nb + 48 + r16) * KDIM + koff;

#pragma unroll 4
    for (int k = 0; k < KDIM; k += 4) {
        v2f a  = *(const v2f*)(Ap  + k);
        v2f b0 = *(const v2f*)(Wp0 + k);
        v2f b1 = *(const v2f*)(Wp1 + k);
        v2f b2 = *(const v2f*)(Wp2 + k);
        v2f b3 = *(const v2f*)(Wp3 + k);
        c0 = wmma_f32(a, b0, c0);
        c1 = wmma_f32(a, b1, c1);
        c2 = wmma_f32(a, b2, c2);
        c3 = wmma_f32(a, b3, c3);
    }

    if (EPI == EPI_SOFTMAX) {  // each 16-col group == one head's 16 logits
        c0 = softmax16(c0); c1 = softmax16(c1); c2 = softmax16(c2); c3 = softmax16(c3);
    }

    if (EPI == EPI_VALUE) {
        // A 16-row tile never crosses a batch boundary (LQ % 16 == 0), so
        // n / lin0 are wave-uniform scalars -- no per-element division.
        const int n    = m0 / LQ;
        const int lin0 = m0 - n * LQ;
        const int col0 = nb + r16;
#pragma unroll
        for (int i = 0; i < 8; ++i) {
            const int lin = lin0 + i + (hi << 3);
            // out[(n, h, lin, ch)] with h = col>>5, ch = col&31
#define WRV(cj, cofs) { const int col = col0 + (cofs);                               \
            out[(((size_t)(n * NHEADS + (col >> 5))) * LQ + lin) * CHN + (col & 31)] = cj[i]; }
            WRV(c0, 0) WRV(c1, 16) WRV(c2, 32) WRV(c3, 48)
#undef WRV
        }
    } else {
#pragma unroll
        for (int i = 0; i < 8; ++i) {
            const int row = m0 + i + (hi << 3);
            float* orow = out + (size_t)row * NC + nb + r16;
            orow[ 0] = c0[i];
            orow[16] = c1[i];
            orow[32] = c2[i];
            orow[48] = c3[i];
        }
    }
}

// One wave per (n, q, h); lane = channel. All position/weight math is
// wave-uniform (broadcast loads). Bilinear corners are branchless: indices
// clamped (load always legal + coalesced), out-of-range weight zeroed via
// cndmask, so all 4 corner gathers per point issue back-to-back.
__global__ __launch_bounds__(256, 4)
void msda_sample(const float* __restrict__ ref,    // (N, Lq, 4, 2)
                 const float* __restrict__ off,    // (N*Lq, 256)  [h][l][p][2]
                 const float* __restrict__ attn,   // (N*Lq, 128)  [h][l*4+p]
                 const float* __restrict__ value,  // (N, H, Lin, 32)
                 float* __restrict__ out)          // (N*Lq, 256)  [h][ch]
{
    const int ch = threadIdx.x;   // 0..31
    const int h  = threadIdx.y;   // 0..7
    const int b  = blockIdx.x;    // n*Lq + q
    const int n  = b / LQ;

    const float* vhead = value + (size_t)(n * NHEADS + h) * LQ * CHN;
    const float* offp  = off  + (size_t)b * 256 + h * 32;
    const float* attnp = attn + (size_t)b * 128 + h * 16;
    const float* refp  = ref  + (size_t)b * 8;

    const int lw[4] = {64, 32, 16, 8};          // square levels
    const int ls[4] = {0, 4096, 5120, 5376};    // level start rows

    float acc = 0.0f;
#pragma unroll
    for (int l = 0; l < 4; ++l) {
        const int   Wi = lw[l];
        const int   Hi = lw[l];
        const float rx = refp[l * 2 + 0];
        const float ry = refp[l * 2 + 1];
        const float* vl = vhead + (size_t)ls[l] * CHN;
#pragma unroll
        for (int p = 0; p < 4; ++p) {
            const float ox = offp[(l * 4 + p) * 2 + 0];
            const float oy = offp[(l * 4 + p) * 2 + 1];
            const float aw = attnp[l * 4 + p];
            // loc = ref + off/(W,H); x = loc_x*W - 0.5 = ref_x*W + off_x - 0.5
            const float x = rx * (float)Wi + ox - 0.5f;
            const float y = ry * (float)Hi + oy - 0.5f;
            const float x0f = floorf(x);
            const float y0f = floorf(y);
            const float wx = x - x0f;
            const float wy = y - y0f;
            const int x0 = (int)x0f;
            const int y0 = (int)y0f;
            const int x1 = x0 + 1;
            const int y1 = y0 + 1;

            // per-axis validity -> weight masks (branchless)
            const float vx0 = (x0 >= 0 && x0 < Wi) ? 1.0f : 0.0f;
            const float vx1 = (x1 >= 0 && x1 < Wi) ? 1.0f : 0.0f;
            const float vy0 = (y0 >= 0 && y0 < Hi) ? 1.0f : 0.0f;
            const float vy1 = (y1 >= 0 && y1 < Hi) ? 1.0f : 0.0f;
            const int cx0 = min(max(x0, 0), Wi - 1);
            const int cx1 = min(max(x1, 0), Wi - 1);
            const int cy0 = min(max(y0, 0), Hi - 1);
            const int cy1 = min(max(y1, 0), Hi - 1);

            const float w00 = (1.0f - wx) * (1.0f - wy) * vx0 * vy0;
            const float w10 = wx * (1.0f - wy) * vx1 * vy0;
            const float w01 = (1.0f - wx) * wy * vx0 * vy1;
            const float w11 = wx * wy * vx1 * vy1;

            const float v00 = vl[((size_t)cy0 * Wi + cx0) * CHN + ch];
            const float v10 = vl[((size_t)cy0 * Wi + cx1) * CHN + ch];
            const float v01 = vl[((size_t)cy1 * Wi + cx0) * CHN + ch];
            const float v11 = vl[((size_t)cy1 * Wi + cx1) * CHN + ch];

            acc += aw * (w00 * v00 + w10 * v10 + w01 * v01 + w11 * v11);
        }
    }
    out[(size_t)b * 256 + h * 32 + ch] = acc;
}

extern "C" void kernel_launch(void* const* d_in, const int* in_sizes, int n_in,
                              void* d_out, int out_size, void* d_ws, size_t ws_size,
                              hipStream_t stream) {
    (void)in_sizes; (void)n_in; (void)out_size; (void)ws_size;

    const float* query  = (const float*)d_in[0];   // (8, 5440, 256)
    const float* refpts = (const float*)d_in[1];   // (8, 5440, 4, 2)
    const float* inflat = (const float*)d_in[2];   // (8, 5440, 256)
    // d_in[3], d_in[4]: spatial shapes / level starts (int32) — baked in.
    const float* W_off  = (const float*)d_in[5];   // (256, 256)
    const float* b_off  = (const float*)d_in[6];
    const float* W_attn = (const float*)d_in[7];   // (128, 256)
    const float* b_attn = (const float*)d_in[8];
    const float* W_val  = (const float*)d_in[9];   // (256, 256)
    const float* b_val  = (const float*)d_in[10];
    const float* W_out  = (const float*)d_in[11];  // (256, 256)
    const float* b_out  = (const float*)d_in[12];
    float* out = (float*)d_out;

    // Workspace carve-up (floats): value 11.14M, off 11.14M, attn 5.57M, tmp 11.14M
    float* ws_value = (float*)d_ws;
    float* ws_off   = ws_value + (size_t)11141120;
    float* ws_attn  = ws_off   + (size_t)11141120;
    float* ws_tmp   = ws_attn  + (size_t)5570560;

    const dim3 blk(256);
    const dim3 grid256(MROWS / 128, 256 / 64);   // (340, 4)
    const dim3 grid128(MROWS / 128, 128 / 64);   // (340, 2)

    // 1) value = input_flatten @ W_val^T + b_val  -> permuted (n,h,lin,32)
    gemm_wmma_f32<256, EPI_VALUE><<<grid256, blk, 0, stream>>>(inflat, W_val, b_val, ws_value);
    // 2) off = query @ W_off^T + b_off
    gemm_wmma_f32<256, EPI_NONE><<<grid256, blk, 0, stream>>>(query, W_off, b_off, ws_off);
    // 3) attn = softmax16(query @ W_attn^T + b_attn), softmax fused in-register
    gemm_wmma_f32<128, EPI_SOFTMAX><<<grid128, blk, 0, stream>>>(query, W_attn, b_attn, ws_attn);
    // 4) deformable bilinear sampling + attention-weighted sum
    msda_sample<<<dim3(MROWS), dim3(32, 8), 0, stream>>>(refpts, ws_off, ws_attn, ws_value, ws_tmp);
    // 5) out = sampled @ W_out^T + b_out
    gemm_wmma_f32<256, EPI_NONE><<<grid256, blk, 0, stream>>>(ws_tmp, W_out, b_out, out);
}